// StabilityLossPoS_57672820850810
// MI455X (gfx1250) — compile-verified
//
#include <hip/hip_runtime.h>
#include <math.h>

// ---------------------------------------------------------------------------
// Stability loss (point-of-support) for MI455X / gfx1250 (wave32, WMMA).
//
// Phase A (per batch): S_mx = sum(mask * x), S_m = sum(mask)
//   mask = (h2/(h2+rho2) < 0.1), h = vertex y
// Phase B (per batch): S_ax = sum(area * centroid_x), S_a = sum(area)
//   area = |cross(p1-p0, p2-p1)|/2, centroid_x = (x0+x1+x2)/3
// loss_b = | S_ax/S_a - S_mx/S_m | ;  out = sum_b loss_b
// ---------------------------------------------------------------------------

#define GMOF_RHO2     (0.02f * 0.02f)
#define CONTACT_THR   0.1f

constexpr int BATCH  = 1024;
constexpr int NV     = 6890;
constexpr int NF     = 13776;
constexpr int TPB    = 256;        // 8 waves of 32
constexpr int NWAVES = TPB / 32;

typedef __attribute__((ext_vector_type(2))) float v2f;
typedef __attribute__((ext_vector_type(8))) float v8f;

// Full-wave (32-lane) sum using the matrix pipe.
// A (16x4 f32) holds per-lane partials: lane L vgpr0 = x_L, vgpr1 = 0.
// With B = all-ones (layout-independent), D[m,n] = x(lane m) + x(lane m+16).
// D layout: vgpr j = row j (lanes 0-15) / row j+8 (lanes 16-31), replicated
// over n.  Sum of the 8 D vgprs gives half-sums; shfl_xor(16) completes it.
__device__ __forceinline__ float wave_reduce_wmma(float x) {
    v2f a; a[0] = x;   a[1] = 0.0f;
    v2f b; b[0] = 1.0f; b[1] = 1.0f;
    v8f c = {};
    c = __builtin_amdgcn_wmma_f32_16x16x4_f32(
            /*neg_a=*/false, a, /*neg_b=*/false, b,
            /*c_mod=*/(short)0, c, /*reuse_a=*/false, /*reuse_b=*/false);
    float t = 0.0f;
#pragma unroll
    for (int j = 0; j < 8; ++j) t += c[j];
    t += __shfl_xor(t, 16, 32);
    return t;   // full 32-lane sum, valid in every lane
}

// Block-wide sum (all 256 threads must call; result returned to every
// thread after the final __syncthreads()).
__device__ __forceinline__ float block_reduce(float x, float* red) {
    float w = wave_reduce_wmma(x);
    const int lane = threadIdx.x & 31;
    const int wv   = threadIdx.x >> 5;
    if (lane == 0) red[wv] = w;
    __syncthreads();
    float t = 0.0f;
#pragma unroll
    for (int i = 0; i < NWAVES; ++i) t += red[i];
    __syncthreads();   // red[] reusable afterwards
    return t;          // same value in every thread
}

__global__ void __launch_bounds__(TPB)
stability_batch_kernel(const float* __restrict__ verts,
                       const int*   __restrict__ faces,
                       float*       __restrict__ batch_loss) {
    __shared__ float red[NWAVES];

    const int b   = blockIdx.x;
    const int tid = threadIdx.x;
    const float* vb = verts + (size_t)b * (size_t)(NV * 3);

    // ---------------- Phase A: contact support (x only needed) -------------
    float s_mx = 0.0f, s_m = 0.0f;
    for (int v = tid; v < NV; v += TPB) {
        const float x  = vb[3 * v + 0];
        const float y  = vb[3 * v + 1];
        const float h2 = y * y;
        const float r  = h2 / (h2 + GMOF_RHO2);
        const float m  = (r < CONTACT_THR) ? 1.0f : 0.0f;
        s_mx += m * x;
        s_m  += m;
    }
    const float S_mx = block_reduce(s_mx, red);
    const float S_m  = block_reduce(s_m,  red);

    // ---------------- Phase B: area-weighted COM (x only needed) -----------
    // Per-batch gather working set (83 KB) is L2-resident (192 MB L2).
    float s_ax = 0.0f, s_a = 0.0f;
    for (int f = tid; f < NF; f += TPB) {
        // speculative prefetch of the next face-index triple (OOB-safe)
        __builtin_prefetch(&faces[3 * (f + TPB)], 0, 0);

        const int i0 = faces[3 * f + 0];
        const int i1 = faces[3 * f + 1];
        const int i2 = faces[3 * f + 2];
        const float* p0 = vb + 3 * i0;
        const float* p1 = vb + 3 * i1;
        const float* p2 = vb + 3 * i2;

        const float x0 = p0[0], y0 = p0[1], z0 = p0[2];
        const float x1 = p1[0], y1 = p1[1], z1 = p1[2];
        const float x2 = p2[0], y2 = p2[1], z2 = p2[2];

        const float ax = x1 - x0, ay = y1 - y0, az = z1 - z0;   // p1 - p0
        const float bx = x2 - x1, by = y2 - y1, bz = z2 - z1;   // p2 - p1

        const float cx = ay * bz - az * by;
        const float cy = az * bx - ax * bz;
        const float cz = ax * by - ay * bx;

        const float area  = 0.5f * sqrtf(cx * cx + cy * cy + cz * cz);
        const float centx = (x0 + x1 + x2) * (1.0f / 3.0f);

        s_ax += area * centx;
        s_a  += area;
    }
    const float S_ax = block_reduce(s_ax, red);
    const float S_a  = block_reduce(s_a,  red);

    if (tid == 0) {
        const float support = S_mx / S_m;
        const float comx    = S_ax / S_a;
        batch_loss[b] = fabsf(comx - support);
    }
}

__global__ void __launch_bounds__(TPB)
final_reduce_kernel(const float* __restrict__ batch_loss,
                    float*       __restrict__ out) {
    __shared__ float red[NWAVES];
    const int tid = threadIdx.x;
    float s = 0.0f;
    for (int i = tid; i < BATCH; i += TPB) s += batch_loss[i];
    const float t = block_reduce(s, red);
    if (tid == 0) out[0] = t;
}

extern "C" void kernel_launch(void* const* d_in, const int* in_sizes, int n_in,
                              void* d_out, int out_size, void* d_ws, size_t ws_size,
                              hipStream_t stream) {
    const float* verts = (const float*)d_in[0];
    const int*   faces = (const int*)d_in[1];   // integer input -> const int*
    float* batch_loss  = (float*)d_ws;          // 1024 floats of scratch
    float* out         = (float*)d_out;

    stability_batch_kernel<<<BATCH, TPB, 0, stream>>>(verts, faces, batch_loss);
    final_reduce_kernel<<<1, TPB, 0, stream>>>(batch_loss, out);
}